// GNNRegressor_47605417509207
// MI455X (gfx1250) — compile-verified
//
#include <hip/hip_runtime.h>
#include <hip/hip_bf16.h>

// ---------------------------------------------------------------------------
// GCN (2x GCNConv + FC) for MI455X / gfx1250.
// fp32 GEMMs via V_WMMA_F32_16X16X4_F32 (exact, reference-matching).
// Edge aggregation via coalesced wave-per-edge gather + global_atomic_add_f32.
// L2 (192MB) holds a1/a2 accumulators -> atomic RMW stays on-chip.
// ---------------------------------------------------------------------------

typedef __attribute__((ext_vector_type(2))) float v2f;
typedef __attribute__((ext_vector_type(8))) float v8f;

static __device__ __forceinline__ int imin(int a, int b) { return a < b ? a : b; }

// ---------------- degree / normalization --------------------------------

__global__ void deg_init_kernel(float* __restrict__ deg, int n) {
    int i = blockIdx.x * blockDim.x + threadIdx.x;
    if (i < n) deg[i] = 1.0f;  // self-loop contributes 1
}

__global__ void deg_count_kernel(const int* __restrict__ dst, float* __restrict__ deg, int E) {
    int e = blockIdx.x * blockDim.x + threadIdx.x;
    if (e < E) atomicAdd(&deg[dst[e]], 1.0f);
}

__global__ void dinv_kernel(float* __restrict__ deg, int n) {
    int i = blockIdx.x * blockDim.x + threadIdx.x;
    if (i < n) deg[i] = rsqrtf(deg[i]);  // deg >= 1 always (self-loop)
}

// ---------------- fp32 WMMA GEMM:  out[n,NOUT] = f(A[n,K]) @ W[K,NOUT] ----
// One wave computes a 16x16 tile, accumulating K in steps of 4 with
// V_WMMA_F32_16X16X4_F32. Block = 256 threads = 8 waves = MWAVES x NWAVES.
// FUSE_RELU_BIAS applies relu(A + bias[k]) on the A-operand load (fuses the
// previous layer's bias+activation for free).

template<int K, int NOUT, int MWAVES, int NWAVES, bool FUSE_RELU_BIAS>
__global__ __launch_bounds__(256) void gemm_wmma_kernel(
    const float* __restrict__ A, const float* __restrict__ W,
    const float* __restrict__ bias, float* __restrict__ out, int n) {

    const int lane = threadIdx.x & 31;
    const int wave = threadIdx.x >> 5;
    const int wn = wave % NWAVES;            // 16-col subtile
    const int wm = wave / NWAVES;            // 16-row subtile
    const int rowBase = blockIdx.x * (MWAVES * 16) + wm * 16;
    const int colBase = wn * 16;
    const int hs  = lane >> 4;               // half-select (K pairing)
    const int l15 = lane & 15;

    // A-operand row for this lane (clamped; tail stores are guarded below,
    // so clamped garbage never reaches memory, and EXEC stays all-1 at WMMA).
    const int arow = imin(rowBase + l15, n - 1);
    const float* Arow = A + (size_t)arow * K;

    v8f acc = {};
#pragma unroll
    for (int k = 0; k < K; k += 4) {
        const int k0 = k + hs * 2;
        float a0 = Arow[k0];
        float a1 = Arow[k0 + 1];
        if (FUSE_RELU_BIAS) {
            a0 = fmaxf(a0 + bias[k0], 0.0f);
            a1 = fmaxf(a1 + bias[k0 + 1], 0.0f);
        }
        v2f a; a.x = a0; a.y = a1;
        v2f b;
        b.x = W[(size_t)k0 * NOUT + colBase + l15];
        b.y = W[(size_t)(k0 + 1) * NOUT + colBase + l15];
        // (neg_a, A, neg_b, B, c_mod, C, reuse_a, reuse_b)
        acc = __builtin_amdgcn_wmma_f32_16x16x4_f32(
            false, a, false, b, (short)0, acc, false, false);
    }

    // C/D layout: VGPR r -> M=r (lanes 0-15), M=r+8 (lanes 16-31); N = lane&15
#pragma unroll
    for (int r = 0; r < 8; ++r) {
        const int row = rowBase + r + hs * 8;
        if (row < n) out[(size_t)row * NOUT + colBase + l15] = acc[r];
    }
}

// ---------------- self-loop init: a[i,:] = h[i,:] * dinv[i]^2 -------------

template<int C>
__global__ void selfloop_kernel(const float* __restrict__ h, const float* __restrict__ dinv,
                                float* __restrict__ a, int n) {
    size_t i = (size_t)blockIdx.x * blockDim.x + threadIdx.x;
    if (i < (size_t)n * C) {
        const int node = (int)(i / C);
        const float d = dinv[node];
        a[i] = h[i] * (d * d);
    }
}

// ---------------- edge aggregation, 64 channels ---------------------------
// One wave per edge: lane l handles channels 2l, 2l+1 (float2 -> one 256B
// fully-coalesced row gather), then two f32 global atomic adds.

__global__ void edge_agg64_kernel(const float* __restrict__ h, const float* __restrict__ dinv,
                                  const int* __restrict__ src, const int* __restrict__ dst,
                                  float* __restrict__ out, int E) {
    const int lane = threadIdx.x & 31;
    const int e = (int)(((size_t)blockIdx.x * blockDim.x + threadIdx.x) >> 5);
    if (e >= E) return;
    const int s = src[e];
    const int d = dst[e];
    const float coef = dinv[s] * dinv[d];
    const float2 v = ((const float2*)(h + (size_t)s * 64))[lane];
    float* op = out + (size_t)d * 64 + lane * 2;
    atomicAdd(op,     v.x * coef);
    atomicAdd(op + 1, v.y * coef);
}

// ---------------- edge aggregation, 32 channels ---------------------------

__global__ void edge_agg32_kernel(const float* __restrict__ h, const float* __restrict__ dinv,
                                  const int* __restrict__ src, const int* __restrict__ dst,
                                  float* __restrict__ out, int E) {
    const int lane = threadIdx.x & 31;
    const int e = (int)(((size_t)blockIdx.x * blockDim.x + threadIdx.x) >> 5);
    if (e >= E) return;
    const int s = src[e];
    const int d = dst[e];
    const float coef = dinv[s] * dinv[d];
    const float v = h[(size_t)s * 32 + lane];
    atomicAdd(out + (size_t)d * 32 + lane, v * coef);
}

// ---------------- FC head: out[i] = relu(a2[i,:]+b2) . Wfc + bfc ----------
// Wave per node, lane per channel, xor-shuffle tree reduction (wave32).

__global__ void fc_kernel(const float* __restrict__ a2, const float* __restrict__ b2,
                          const float* __restrict__ Wfc, const float* __restrict__ bfc,
                          float* __restrict__ out, int n) {
    const int lane = threadIdx.x & 31;
    const int node = (int)(((size_t)blockIdx.x * blockDim.x + threadIdx.x) >> 5);
    if (node >= n) return;
    float v = fmaxf(a2[(size_t)node * 32 + lane] + b2[lane], 0.0f) * Wfc[lane];
#pragma unroll
    for (int off = 16; off > 0; off >>= 1) v += __shfl_xor(v, off, 32);
    if (lane == 0) out[node] = v + bfc[0];
}

// ---------------- host-side orchestration --------------------------------

extern "C" void kernel_launch(void* const* d_in, const int* in_sizes, int n_in,
                              void* d_out, int out_size, void* d_ws, size_t ws_size,
                              hipStream_t stream) {
    const float* x   = (const float*)d_in[0];   // [n,128]
    const int*   ei  = (const int*)  d_in[1];   // [2,E] (int32; JAX default x64-off)
    const float* W1  = (const float*)d_in[2];   // [128,64]
    const float* b1  = (const float*)d_in[3];   // [64]
    const float* W2  = (const float*)d_in[4];   // [64,32]
    const float* b2  = (const float*)d_in[5];   // [32]
    const float* Wfc = (const float*)d_in[6];   // [32,1]
    const float* bfc = (const float*)d_in[7];   // [1]

    const int n = in_sizes[0] / 128;
    const int E = in_sizes[1] / 2;
    const int* srcp = ei;
    const int* dstp = ei + E;

    // Workspace layout (reuse: h2 aliases h1, a2 aliases a1). Peak ~52MB.
    float* dinv = (float*)d_ws;                          // n
    float* h1   = dinv + (((size_t)n + 255) & ~(size_t)255);   // n*64
    float* a1   = h1 + (size_t)n * 64;                   // n*64
    float* h2   = h1;                                    // n*32 (h1 dead)
    float* a2   = a1;                                    // n*32 (a1 dead)

    const int T = 256;
    // 1) degrees -> dinv
    deg_init_kernel<<<(n + T - 1) / T, T, 0, stream>>>(dinv, n);
    deg_count_kernel<<<(E + T - 1) / T, T, 0, stream>>>(dstp, dinv, E);
    dinv_kernel<<<(n + T - 1) / T, T, 0, stream>>>(dinv, n);

    // 2) h1 = x @ W1   (128->64), 2 M-waves x 4 N-waves => 32 rows/block
    gemm_wmma_kernel<128, 64, 2, 4, false>
        <<<(n + 31) / 32, 256, 0, stream>>>(x, W1, nullptr, h1, n);

    // 3) a1 = selfloop + scatter-add over edges (64 ch)
    {
        size_t tot = (size_t)n * 64;
        selfloop_kernel<64><<<(unsigned)((tot + T - 1) / T), T, 0, stream>>>(h1, dinv, a1, n);
        size_t waves = (size_t)E * 32;
        edge_agg64_kernel<<<(unsigned)((waves + T - 1) / T), T, 0, stream>>>(h1, dinv, srcp, dstp, a1, E);
    }

    // 4) h2 = relu(a1 + b1) @ W2   (64->32), 4 M-waves x 2 N-waves => 64 rows/block
    gemm_wmma_kernel<64, 32, 4, 2, true>
        <<<(n + 63) / 64, 256, 0, stream>>>(a1, W2, b1, h2, n);

    // 5) a2 = selfloop + scatter-add over edges (32 ch)
    {
        size_t tot = (size_t)n * 32;
        selfloop_kernel<32><<<(unsigned)((tot + T - 1) / T), T, 0, stream>>>(h2, dinv, a2, n);
        size_t waves = (size_t)E * 32;
        edge_agg32_kernel<<<(unsigned)((waves + T - 1) / T), T, 0, stream>>>(h2, dinv, srcp, dstp, a2, E);
    }

    // 6) out = relu(a2 + b2) @ Wfc + bfc
    {
        size_t waves = (size_t)n * 32;
        fc_kernel<<<(unsigned)((waves + T - 1) / T), T, 0, stream>>>(a2, b2, Wfc, bfc, (float*)d_out, n);
    }
}